// GAT_13073880449095
// MI455X (gfx1250) — compile-verified
//
#include <hip/hip_runtime.h>

#define N_NODES 4096
#define ALPHA_SLOPE 0.2f
#define NEG_BIG_F (-9e15f)
#define KSPLIT 4

typedef float v2f __attribute__((ext_vector_type(2)));
typedef float v4f __attribute__((ext_vector_type(4)));
typedef float v8f __attribute__((ext_vector_type(8)));

#define WMMA_F32(a, b, c) \
  __builtin_amdgcn_wmma_f32_16x16x4_f32(false, (a), false, (b), (short)0, (c), false, false)

// ---------------------------------------------------------------------------
// Generic fp32 WMMA GEMM: C[M,N] = A[M,K] @ B[K,N]  (row-major)
// Block 256 threads (8 waves), tile 128(M) x 64(N), K-chunk 16,
// double-buffered LDS: next chunk's global loads issue before this chunk's
// WMMAs; one barrier per chunk. Requires Mdim%128==0, Kdim even, Ndim%4==0.
// ---------------------------------------------------------------------------
__global__ __launch_bounds__(256) void wmma_gemm_f32(
    const float* __restrict__ A, const float* __restrict__ B,
    float* __restrict__ C, int Mdim, int Ndim, int Kdim) {
  __shared__ float As[2][128][20];
  __shared__ float Bs[2][16][68];

  const int tid  = threadIdx.x;
  const int lane = tid & 31, wid = tid >> 5;
  const int half = lane >> 4, l15 = lane & 15;
  const int wm = wid & 3, wn = wid >> 2;
  const int bM = blockIdx.x * 128, bN = blockIdx.y * 64;

  const v2f* __restrict__ A2 = (const v2f*)A;
  const v4f* __restrict__ B4 = (const v4f*)B;

  // fill coordinates (loop-invariant)
  const int arow = tid >> 3, ac2 = (tid & 7) * 2;   // A: rows arow+{0,32,64,96}
  const int brow = tid >> 4, bc4 = (tid & 15) * 4;  // B: 1 float4 / thread
  const int gn = bN + bc4;
  const int gnc = min(gn, Ndim - 4);
  const bool gnok = gn < Ndim;

  v2f arg[4];
  v4f brg;

  auto loadTiles = [&](int kt) {
    const int gkc = min(kt + ac2, Kdim - 2);
#pragma unroll
    for (int t = 0; t < 4; ++t)
      arg[t] = A2[((long)(bM + arow + t * 32) * Kdim + gkc) >> 1];
    const int gkb = min(kt + brow, Kdim - 1);
    brg = B4[((long)gkb * Ndim + gnc) >> 2];
  };

  auto storeTiles = [&](int buf, int kt) {
    const bool ok0 = (kt + ac2) < Kdim, ok1 = (kt + ac2 + 1) < Kdim;
#pragma unroll
    for (int t = 0; t < 4; ++t) {
      v2f v = arg[t];
      v.x = ok0 ? v.x : 0.0f;
      v.y = ok1 ? v.y : 0.0f;
      *(v2f*)&As[buf][arow + t * 32][ac2] = v;
    }
    v4f bv = brg;
    if (!(((kt + brow) < Kdim) && gnok)) { bv.x = 0.f; bv.y = 0.f; bv.z = 0.f; bv.w = 0.f; }
    *(v4f*)&Bs[buf][brow][bc4] = bv;
  };

  v8f acc[2][2] = {};

  loadTiles(0);
  storeTiles(0, 0);
  __syncthreads();

  int cur = 0;
  for (int kt = 0; kt < Kdim; kt += 16) {
    const int ktn = kt + 16;
    const bool hasNext = ktn < Kdim;
    if (hasNext) loadTiles(ktn);   // global loads in flight during WMMAs

#pragma unroll
    for (int kk = 0; kk < 16; kk += 4) {
      const int k2 = kk + 2 * half;  // K order: v0/lo=K0, v1/lo=K1, v0/hi=K2, v1/hi=K3
      v2f a[2], b[2];
#pragma unroll
      for (int i = 0; i < 2; ++i) {
        int m = wm * 32 + i * 16 + l15;
        a[i].x = As[cur][m][k2];
        a[i].y = As[cur][m][k2 + 1];
      }
#pragma unroll
      for (int j = 0; j < 2; ++j) {
        int n = wn * 32 + j * 16 + l15;
        b[j].x = Bs[cur][k2][n];
        b[j].y = Bs[cur][k2 + 1][n];
      }
#pragma unroll
      for (int i = 0; i < 2; ++i)
#pragma unroll
        for (int j = 0; j < 2; ++j)
          acc[i][j] = WMMA_F32(a[i], b[j], acc[i][j]);
    }

    if (hasNext) {
      storeTiles(cur ^ 1, ktn);
      __syncthreads();
      cur ^= 1;
    }
  }

#pragma unroll
  for (int i = 0; i < 2; ++i)
#pragma unroll
    for (int j = 0; j < 2; ++j) {
      int col = bN + wn * 32 + j * 16 + l15;
      if (col < Ndim) {
        for (int r = 0; r < 8; ++r) {
          int row = bM + wm * 32 + i * 16 + 8 * half + r;
          C[(long)row * Ndim + col] = acc[i][j][r];
        }
      }
    }
}

// ---------------------------------------------------------------------------
// f_self / f_neigh: two length-D dot products per row (wave per row, b128)
// ---------------------------------------------------------------------------
__global__ __launch_bounds__(256) void row_dots(
    const float* __restrict__ H, const float* __restrict__ a_s,
    const float* __restrict__ a_n, float* __restrict__ fs,
    float* __restrict__ fn, int D) {
  const int wid = threadIdx.x >> 5, lane = threadIdx.x & 31;
  const int row = blockIdx.x * 8 + wid;
  const v4f* __restrict__ H4 = (const v4f*)H;
  const v4f* __restrict__ S4 = (const v4f*)a_s;
  const v4f* __restrict__ N4 = (const v4f*)a_n;
  float s = 0.0f, n = 0.0f;
  for (int d4 = lane * 4; d4 < D; d4 += 128) {
    v4f v = H4[((long)row * D + d4) >> 2];
    v4f sa = S4[d4 >> 2], na = N4[d4 >> 2];
    s += v.x * sa.x + v.y * sa.y + v.z * sa.z + v.w * sa.w;
    n += v.x * na.x + v.y * na.y + v.z * na.z + v.w * na.w;
  }
  for (int off = 16; off > 0; off >>= 1) {
    s += __shfl_down(s, off);
    n += __shfl_down(n, off);
  }
  if (lane == 0) { fs[row] = s; fn[row] = n; }
}

// ---------------------------------------------------------------------------
// Global max of fn[0..4095] -> fnmax[0]   (single block)
// ---------------------------------------------------------------------------
__global__ __launch_bounds__(256) void vec_max(const float* __restrict__ v,
                                               float* __restrict__ out) {
  __shared__ float red[256];
  float m = -3.4e38f;
  for (int i = threadIdx.x; i < N_NODES; i += 256) m = fmaxf(m, v[i]);
  red[threadIdx.x] = m;
  __syncthreads();
  for (int s = 128; s > 0; s >>= 1) {
    if (threadIdx.x < s) red[threadIdx.x] = fmaxf(red[threadIdx.x], red[threadIdx.x + s]);
    __syncthreads();
  }
  if (threadIdx.x == 0) out[0] = red[0];
}

// ---------------------------------------------------------------------------
// Single-pass static-bound flash att@h (unnormalized), K split 4 ways.
//   p_ij = exp(e_ij - b_i),  b_i = max(0, fs_i + max_j fn_j)  >= true rowmax
//   pOut[s] = p * H  (WMMA),  pL[s][i] = sum_j p_ij     (s = K-split)
// Block 256 threads: tile 64(M) x TN=64*NT(N), K-chunk 16, wave grid 2x4.
// Double-buffered LDS; exp math co-executes with the WMMA block.
// M/adj are read exactly once across all splits; no materialized attention.
// ---------------------------------------------------------------------------
template <int NT>
__global__ __launch_bounds__(256) void att_flash(
    const float* __restrict__ Mm, const float* __restrict__ adj,
    const float* __restrict__ fs, const float* __restrict__ fn,
    const float* __restrict__ fnmaxp, const float* __restrict__ H,
    float* __restrict__ pOut, float* __restrict__ pL, int Ndim) {
  constexpr int TN = 64 * NT;
  __shared__ float As[2][64][20];
  __shared__ float Bs[2][16][TN + 4];

  const int tid  = threadIdx.x;
  const int lane = tid & 31, wid = tid >> 5;
  const int half = lane >> 4, l15 = lane & 15;
  const int wm = wid & 1, wn = wid >> 1;
  const int bM = blockIdx.x * 64;
  const int kBeg = blockIdx.y * (N_NODES / KSPLIT);
  const int kEnd = kBeg + (N_NODES / KSPLIT);

  const int ar  = tid >> 2;        // att row this thread fills
  const int ac4 = (tid & 3) * 4;   // column-of-chunk base (float4)
  const float fsr = fs[bM + ar];
  const float bnd = fmaxf(0.0f, fsr + fnmaxp[0]);
  float lacc = 0.0f;

  // B fill coordinates (loop-invariant)
  const int br0 = tid / (TN / 4);
  const int bc4 = (tid % (TN / 4)) * 4;
  const int bgc = min(bc4, Ndim - 4);
  const bool bok = bc4 < Ndim;

  const v4f* __restrict__ Mm4 = (const v4f*)Mm;
  const v4f* __restrict__ Ad4 = (const v4f*)adj;
  const v4f* __restrict__ Fn4 = (const v4f*)fn;
  const v4f* __restrict__ H4  = (const v4f*)H;

  v4f m4, a4, f4, brg[NT];

  auto loadTiles = [&](int kt) {
    long g4 = ((long)(bM + ar) * N_NODES + kt + ac4) >> 2;
    m4 = Mm4[g4];
    a4 = Ad4[g4];
    f4 = Fn4[(kt + ac4) >> 2];
#pragma unroll
    for (int t = 0; t < NT; ++t)
      brg[t] = H4[((long)(kt + br0 + t * (16 / NT) * NT) * Ndim + bgc) >> 2];
  };

  auto storeTiles = [&](int buf) {
    v4f p;
#pragma unroll
    for (int e = 0; e < 4; ++e) {
      float ev = (fsr + f4[e]) * m4[e];
      ev = ev > 0.0f ? ev : ALPHA_SLOPE * ev;
      ev = a4[e] > 0.0f ? ev : NEG_BIG_F;
      p[e] = __expf(ev - bnd);
      lacc += p[e];
    }
    *(v4f*)&As[buf][ar][ac4] = p;
#pragma unroll
    for (int t = 0; t < NT; ++t) {
      v4f bv = brg[t];
      if (!bok) { bv.x = 0.f; bv.y = 0.f; bv.z = 0.f; bv.w = 0.f; }
      *(v4f*)&Bs[buf][br0 + t * (16 / NT) * NT][bc4] = bv;
    }
  };

  v8f acc[2][NT] = {};

  loadTiles(kBeg);
  storeTiles(0);
  __syncthreads();

  int cur = 0;
  for (int kt = kBeg; kt < kEnd; kt += 16) {
    const int ktn = kt + 16;
    const bool hasNext = ktn < kEnd;
    if (hasNext) loadTiles(ktn);

#pragma unroll
    for (int kk = 0; kk < 16; kk += 4) {
      const int k2 = kk + 2 * half;
      v2f a[2], b[NT];
#pragma unroll
      for (int i = 0; i < 2; ++i) {
        int m = wm * 32 + i * 16 + l15;
        a[i].x = As[cur][m][k2];
        a[i].y = As[cur][m][k2 + 1];
      }
#pragma unroll
      for (int j = 0; j < NT; ++j) {
        int n = wn * 16 * NT + j * 16 + l15;
        b[j].x = Bs[cur][k2][n];
        b[j].y = Bs[cur][k2 + 1][n];
      }
#pragma unroll
      for (int i = 0; i < 2; ++i)
#pragma unroll
        for (int j = 0; j < NT; ++j)
          acc[i][j] = WMMA_F32(a[i], b[j], acc[i][j]);
    }

    if (hasNext) {
      storeTiles(cur ^ 1);   // exp + LDS stores overlap the tensor pipe drain
      __syncthreads();
      cur ^= 1;
    }
  }

  // partial row-sum of p (reduce across the 4 lanes of each row group)
  float lsum = lacc;
  lsum += __shfl_xor(lsum, 1, 4);
  lsum += __shfl_xor(lsum, 2, 4);
  if ((tid & 3) == 0) pL[(long)blockIdx.y * N_NODES + bM + ar] = lsum;

#pragma unroll
  for (int i = 0; i < 2; ++i)
#pragma unroll
    for (int j = 0; j < NT; ++j) {
      int col = wn * 16 * NT + j * 16 + l15;
      if (col < Ndim) {
        for (int r = 0; r < 8; ++r) {
          int row = bM + wm * 32 + i * 16 + 8 * half + r;
          pOut[((long)blockIdx.y * N_NODES + row) * Ndim + col] = acc[i][j][r];
        }
      }
    }
}

// ---------------------------------------------------------------------------
// Combine K-splits: out = elu( (sum_s pOut[s]) / (sum_s pL[s]) )
// ---------------------------------------------------------------------------
__global__ __launch_bounds__(256) void att_finalize(
    const float* __restrict__ pOut, const float* __restrict__ pL,
    float* __restrict__ out, int Ndim, int rowShift) {
  const int idx = blockIdx.x * 256 + threadIdx.x;
  const int row = idx >> rowShift;
  float s = 0.0f, l = 0.0f;
#pragma unroll
  for (int k = 0; k < KSPLIT; ++k) {
    s += pOut[(long)k * N_NODES * Ndim + idx];
    l += pL[k * N_NODES + row];
  }
  float v = s / l;
  out[idx] = v > 0.0f ? v : (__expf(v) - 1.0f);
}

// ---------------------------------------------------------------------------
// Row-wise L2 normalize (E=16); z -> ws and tail of d_out
// ---------------------------------------------------------------------------
__global__ __launch_bounds__(256) void norm_rows(
    const float* __restrict__ H, float* __restrict__ z1,
    float* __restrict__ z2) {
  const int r = blockIdx.x * 16 + (threadIdx.x >> 4);
  const int c = threadIdx.x & 15;
  float v = H[r * 16 + c];
  float s = v * v;
  for (int off = 8; off > 0; off >>= 1) s += __shfl_xor(s, off, 16);
  float nrm = fmaxf(sqrtf(s), 1e-12f);
  float zv = v / nrm;
  z1[r * 16 + c] = zv;
  z2[r * 16 + c] = zv;
}

// ---------------------------------------------------------------------------
// A_pred = sigmoid(z @ z^T), K=16: pure-register WMMA, one 16x16 tile / wave
// ---------------------------------------------------------------------------
__global__ __launch_bounds__(256) void zzt_sigmoid(
    const float* __restrict__ z, float* __restrict__ out) {
  const int tid  = threadIdx.x;
  const int lane = tid & 31, wid = tid >> 5;
  const int half = lane >> 4, l15 = lane & 15;
  const int tile = blockIdx.x * 8 + wid;  // 65536 tiles
  const int i0 = (tile >> 8) * 16, j0 = (tile & 255) * 16;

  v8f acc = {};
#pragma unroll
  for (int kk = 0; kk < 16; kk += 4) {
    const int k2 = kk + 2 * half;
    v2f a = *(const v2f*)&z[(i0 + l15) * 16 + k2];   // A[m][k]
    v2f b = *(const v2f*)&z[(j0 + l15) * 16 + k2];   // B[k][n] = z[j0+n][k]
    acc = WMMA_F32(a, b, acc);
  }
#pragma unroll
  for (int r = 0; r < 8; ++r) {
    int row = i0 + r + 8 * half;
    out[(long)row * N_NODES + j0 + l15] = 1.0f / (1.0f + __expf(-acc[r]));
  }
}

// ---------------------------------------------------------------------------
extern "C" void kernel_launch(void* const* d_in, const int* in_sizes, int n_in,
                              void* d_out, int out_size, void* d_ws, size_t ws_size,
                              hipStream_t stream) {
  const float* x   = (const float*)d_in[0];   // [4096,1870]
  const float* adj = (const float*)d_in[1];   // [4096,4096]
  const float* Mm  = (const float*)d_in[2];   // [4096,4096]
  const float* W1  = (const float*)d_in[3];   // [1870,256]
  const float* as1 = (const float*)d_in[4];   // [256]
  const float* an1 = (const float*)d_in[5];   // [256]
  const float* W2  = (const float*)d_in[6];   // [256,16]
  const float* as2 = (const float*)d_in[7];   // [16]
  const float* an2 = (const float*)d_in[8];   // [16]
  float* out = (float*)d_out;                 // [4096*4096 A_pred ; 4096*16 z]

  const int Nn = N_NODES, F = 1870, Hd = 256, E = 16;

  float* ws    = (float*)d_ws;
  float* h1    = ws;                              // 4096*256
  float* hout1 = h1    + (long)Nn * Hd;           // 4096*256
  float* h2    = hout1 + (long)Nn * Hd;           // 4096*16
  float* hout2 = h2    + (long)Nn * E;            // 4096*16
  float* fs    = hout2 + (long)Nn * E;            // 4096
  float* fn    = fs + Nn;                         // 4096
  float* fnmx  = fn + Nn;                         // 16 (padded)
  float* pL    = fnmx + 16;                       // KSPLIT*4096
  float* pOut  = pL + KSPLIT * Nn;                // KSPLIT*4096*256
  float* zbuf  = pOut + (long)KSPLIT * Nn * Hd;   // 4096*16

  // ---- layer 1 ----
  wmma_gemm_f32<<<dim3(Nn / 128, Hd / 64), 256, 0, stream>>>(x, W1, h1, Nn, Hd, F);
  row_dots<<<Nn / 8, 256, 0, stream>>>(h1, as1, an1, fs, fn, Hd);
  vec_max<<<1, 256, 0, stream>>>(fn, fnmx);
  att_flash<4><<<dim3(Nn / 64, KSPLIT), 256, 0, stream>>>(
      Mm, adj, fs, fn, fnmx, h1, pOut, pL, Hd);
  att_finalize<<<(Nn * Hd) / 256, 256, 0, stream>>>(pOut, pL, hout1, Hd, 8);

  // ---- layer 2 ----
  wmma_gemm_f32<<<dim3(Nn / 128, 1), 256, 0, stream>>>(hout1, W2, h2, Nn, E, Hd);
  row_dots<<<Nn / 8, 256, 0, stream>>>(h2, as2, an2, fs, fn, E);
  vec_max<<<1, 256, 0, stream>>>(fn, fnmx);
  att_flash<1><<<dim3(Nn / 64, KSPLIT), 256, 0, stream>>>(
      Mm, adj, fs, fn, fnmx, h2, pOut, pL, E);
  att_finalize<<<(Nn * E) / 256, 256, 0, stream>>>(pOut, pL, hout2, E, 4);

  // ---- decoder ----
  norm_rows<<<Nn / 16, 256, 0, stream>>>(hout2, zbuf, out + (long)Nn * Nn);
  zzt_sigmoid<<<(Nn / 16) * (Nn / 16) / 8, 256, 0, stream>>>(zbuf, out);
}